// OctreeMaxUnpool_17377437679940
// MI455X (gfx1250) — compile-verified
//
#include <hip/hip_runtime.h>

// OctreeMaxUnpool scatter: out[(i*8+c)*64 + k] = (indices[i,k]==c) ? data[nempty_idx[i],k] : 0
// Memory-bound (~1.28 GB traffic, ~55us floor @ 23.3 TB/s). No matmul structure ->
// optimize the data-movement path: gfx1250 async global->LDS prefetch pipeline,
// b128 coalesced accesses, non-temporal b128 stores.

typedef float v4f __attribute__((ext_vector_type(4)));
typedef int   v4i __attribute__((ext_vector_type(4)));

#define THREADS 256
#define BLOCKS  2048

__global__ __launch_bounds__(THREADS)
void octree_unpool_async_kernel(const float* __restrict__ data,
                                const int*   __restrict__ indices,
                                const int*   __restrict__ nempty,
                                float*       __restrict__ out,
                                long long total /* num*16 tasks */) {
    // Per-thread private staging: 2 stages x (16B data + 16B indices) = 64B/thread.
    __shared__ char lds[THREADS * 64];
    const int tid = threadIdx.x;
    char* slot = &lds[tid * 64];
    // Generic (flat) pointer to LDS has the wave-relative LDS byte offset in its
    // low 32 bits (shared aperture occupies addr[63:32]); truncate to feed the
    // async-load VDST (LDS address) VGPR.
    const unsigned slot_off = (unsigned)(unsigned long long)slot;

    const long long S = (long long)gridDim.x * THREADS;
    long long g = (long long)blockIdx.x * THREADS + tid;
    if (g >= total) return;

    // Issue async global->LDS loads for task gg into stage st (clamped in-bounds).
    auto issue = [&](long long gg, int st) {
        long long gc = (gg < total) ? gg : (total - 1);
        int i  = (int)(gc >> 4);     // node index
        int ls = (int)(gc & 15);     // 16B chunk within the 256B row
        int row = nempty[i];         // sorted gather index (dependent load)
        unsigned long long gd = (unsigned long long)(data    + (size_t)row * 64 + (size_t)ls * 4);
        unsigned long long gi = (unsigned long long)(indices + (size_t)i   * 64 + (size_t)ls * 4);
        unsigned l0 = slot_off + (unsigned)(st * 32);
        unsigned l1 = l0 + 16u;
        asm volatile("global_load_async_to_lds_b128 %0, %1, off"
                     :: "v"(l0), "v"(gd) : "memory");
        asm volatile("global_load_async_to_lds_b128 %0, %1, off"
                     :: "v"(l1), "v"(gi) : "memory");
    };

    issue(g, 0);                    // prologue: stage 0 in flight
    int stage = 0;
    for (; g < total; g += S) {
        // Prefetch next tile into the other stage, then wait until only those
        // 2 newest async ops remain outstanding (in-order completion => the
        // current stage's 2 loads have landed in LDS).
        issue(g + S, stage ^ 1);
        asm volatile("s_wait_asynccnt 0x2" ::: "memory");

        const v4f d4 = *(const v4f*)(slot + stage * 32);        // ds_load_b128
        const v4i i4 = *(const v4i*)(slot + stage * 32 + 16);   // ds_load_b128

        int i  = (int)(g >> 4);
        int ls = (int)(g & 15);
        float* op = out + (size_t)i * 512 + (size_t)ls * 4;     // row i*8, chunk ls

        #pragma unroll
        for (int c = 0; c < 8; ++c) {
            v4f v;
            v.x = (i4.x == c) ? d4.x : 0.0f;
            v.y = (i4.y == c) ? d4.y : 0.0f;
            v.z = (i4.z == c) ? d4.z : 0.0f;
            v.w = (i4.w == c) ? d4.w : 0.0f;
            // 1 GB output stream > 192 MB L2: bypass with non-temporal b128 store.
            __builtin_nontemporal_store(v, (v4f*)(op + c * 64));
        }
        stage ^= 1;
    }
    // Tail prefetch (clamped, unused) drains via s_endpgm's implicit wait-idle.
}

extern "C" void kernel_launch(void* const* d_in, const int* in_sizes, int n_in,
                              void* d_out, int out_size, void* d_ws, size_t ws_size,
                              hipStream_t stream) {
    const float* data    = (const float*)d_in[0];   // [N, 64] f32
    const int*   indices = (const int*)  d_in[1];   // [num, 64] i32 in [0,8)
    const int*   nempty  = (const int*)  d_in[2];   // [num] i32 sorted
    float*       out     = (float*)d_out;           // [num*8, 64] f32

    const long long num   = (long long)in_sizes[2];
    const long long total = num * 16;               // (node, 4-channel chunk) tasks

    octree_unpool_async_kernel<<<BLOCKS, THREADS, 0, stream>>>(
        data, indices, nempty, out, total);
}